// NN_Pooling_46634754900232
// MI455X (gfx1250) — compile-verified
//
#include <hip/hip_runtime.h>
#include <hip/hip_bf16.h>

typedef __attribute__((ext_vector_type(2))) float v2f;
typedef __attribute__((ext_vector_type(8))) float v8f;

#define NAGENT 4096
#define ROWS_PER_BLOCK 8
#define NTHREADS 256   // 8 wave32 waves

__global__ __launch_bounds__(NTHREADS) void nn_pool_kernel(
    const float* __restrict__ obs1, const float* __restrict__ obs2,
    const float* __restrict__ W, const float* __restrict__ bias,
    float* __restrict__ out)
{
    __shared__ float2 s_pos[NAGENT];        // 32 KB: obs2 staged in LDS
    __shared__ float  s_feats[32][4];       // 32 feature rows x 4 feats

    const int tid  = threadIdx.x;
    const int wave = tid >> 5;
    const int lane = tid & 31;
    const int i0   = blockIdx.x * ROWS_PER_BLOCK;

    const float2* p1 = (const float2*)obs1;
    const float2* p2 = (const float2*)obs2;

    // ---- stage obs2 into LDS (conflict-free b64 stores) ----
    for (int idx = tid; idx < NAGENT; idx += NTHREADS) {
        s_pos[idx] = p2[idx];
    }
    __syncthreads();

    // ---- per-wave: row i, lane-strided scan of all j with running top-4 ----
    const int i = i0 + wave;
    const float2 pi = s_pos[i];
    const float INF = __builtin_inff();
    float d0 = INF, d1 = INF, d2 = INF, d3 = INF;
    int  id0 = -1,  id1 = -1,  id2 = -1,  id3 = -1;

    for (int j = lane; j < NAGENT; j += 32) {
        float2 pj = s_pos[j];
        float dx = pj.x - pi.x;
        float dy = pj.y - pi.y;
        float dd = dx * dx + dy * dy;       // squared dist: same ordering as sqrt
        if (j == i) dd = INF;               // exclude self (matches inf diagonal)
        bool c0 = dd < d0, c1 = dd < d1, c2 = dd < d2, c3 = dd < d3;
        d3 = c2 ? d2 : (c3 ? dd : d3);  id3 = c2 ? id2 : (c3 ? j : id3);
        d2 = c1 ? d1 : (c2 ? dd : d2);  id2 = c1 ? id1 : (c2 ? j : id2);
        d1 = c0 ? d0 : (c1 ? dd : d1);  id1 = c0 ? id0 : (c1 ? j : id1);
        d0 = c0 ? dd : d0;              id0 = c0 ? j  : id0;
    }

    // ---- cross-lane extraction of global top-4 (ordered, index tie-break) ----
    int sel[4];
    #pragma unroll
    for (int r = 0; r < 4; r++) {
        float cd = d0; int cj = id0;
        #pragma unroll
        for (int off = 16; off >= 1; off >>= 1) {
            float od = __shfl_xor(cd, off, 32);
            int   oj = __shfl_xor(cj, off, 32);
            bool take = (od < cd) || (od == cd && oj < cj);
            cd = take ? od : cd;
            cj = take ? oj : cj;
        }
        sel[r] = cj;                        // all lanes agree on round-r winner
        if (id0 == cj) {                    // owning lane pops its front
            d0 = d1; id0 = id1;
            d1 = d2; id1 = id2;
            d2 = d3; id2 = id3;
            d3 = INF; id3 = -1;
        }
    }

    // ---- lanes 0..3 build the 4 feature rows for this agent ----
    if (lane < 4) {
        int j = (lane == 0) ? sel[0] : (lane == 1) ? sel[1]
              : (lane == 2) ? sel[2] : sel[3];
        float2 aj = p1[j], bj = p2[j];      // neighbor obs1/obs2 (tiny, L2-hot)
        float2 ai = p1[i], bi = p2[i];
        int fr = wave * 4 + lane;
        s_feats[fr][0] = bj.x - bi.x;                         // rel_pos.x
        s_feats[fr][1] = bj.y - bi.y;                         // rel_pos.y
        s_feats[fr][2] = (bj.x - aj.x) - (bi.x - ai.x);       // rel_vel.x
        s_feats[fr][3] = (bj.y - aj.y) - (bi.y - ai.y);       // rel_vel.y
    }
    __syncthreads();

    // ---- waves 0/1: f32 WMMA 16x16x4  D = feats(16x4) * W^T(4x16) ----
    if (wave < 2) {
        int m    = lane & 15;               // A-row / B-col / D-col index
        int half = lane >> 4;               // K pair select: 0 -> K=0,1 ; 1 -> K=2,3

        // A fragment: 16x4 f32, lane m holds K = 2*half, 2*half+1 of row m
        v2f a;
        a.x = s_feats[wave * 16 + m][2 * half];
        a.y = s_feats[wave * 16 + m][2 * half + 1];

        // B fragment: B[k][n] = W[n][k], zero-padded for n >= 8 (branchless)
        int   nc  = m & 7;
        float msk = (m < 8) ? 1.0f : 0.0f;
        v2f bf;
        bf.x = W[nc * 4 + 2 * half]     * msk;
        bf.y = W[nc * 4 + 2 * half + 1] * msk;

        v8f c = {};
        c = __builtin_amdgcn_wmma_f32_16x16x4_f32(
                /*neg_a=*/false, a, /*neg_b=*/false, bf,
                /*c_mod=*/(short)0, c, /*reuse_a=*/false, /*reuse_b=*/false);

        // D layout: lane l (N = l%16), VGPR r -> M = r + 8*(l/16)
        float bv = bias[nc];
        if (m < 8) {
            #pragma unroll
            for (int r = 0; r < 8; r++) {
                int M   = r + 8 * half;
                int frr = wave * 16 + M;          // global feature row
                int gi  = i0 + (frr >> 2);        // agent index
                int k   = frr & 3;                // neighbor slot
                float v = c[r] + bv;
                out[gi * 32 + k * 8 + m] = v > 0.0f ? v : 0.0f;
            }
        }
    }
}

extern "C" void kernel_launch(void* const* d_in, const int* in_sizes, int n_in,
                              void* d_out, int out_size, void* d_ws, size_t ws_size,
                              hipStream_t stream) {
    const float* obs1 = (const float*)d_in[0];
    const float* obs2 = (const float*)d_in[1];
    const float* W    = (const float*)d_in[2];
    const float* b    = (const float*)d_in[3];
    float* out = (float*)d_out;
    int n = in_sizes[0] / 2;                 // 4096
    int blocks = n / ROWS_PER_BLOCK;         // 512
    nn_pool_kernel<<<blocks, NTHREADS, 0, stream>>>(obs1, obs2, W, b, out);
}